// BRGEHHNet_81217831567791
// MI455X (gfx1250) — compile-verified
//
#include <hip/hip_runtime.h>
#include <math.h>

#define N_AGENT 25
#define N_S     32
#define N_A     4
#define EHH     800      // N_AGENT * N_S
#define HID     3200     // EHH_HID
#define BATCH   2048
#define N_EDGE  300

typedef float v2f __attribute__((ext_vector_type(2)));
typedef float v8f __attribute__((ext_vector_type(8)));

__device__ __forceinline__ float leaky(float x) { return x >= 0.f ? x : 0.01f * x; }

// ---------------- K1: per-feature mean / rstd over batch ----------------
__global__ void k_meanvar(const float* __restrict__ states,
                          float* __restrict__ mu, float* __restrict__ rstd) {
    int f = blockIdx.x;              // 0..799
    int a = f >> 5, s = f & 31;
    const float* base = states + (size_t)a * BATCH * N_S + s;
    float sum = 0.f, sq = 0.f;
    for (int b = threadIdx.x; b < BATCH; b += blockDim.x) {
        float v = base[b * N_S];
        sum += v; sq += v * v;
    }
    __shared__ float ssum[256], ssq[256];
    ssum[threadIdx.x] = sum; ssq[threadIdx.x] = sq;
    __syncthreads();
    for (int off = 128; off > 0; off >>= 1) {
        if ((int)threadIdx.x < off) {
            ssum[threadIdx.x] += ssum[threadIdx.x + off];
            ssq[threadIdx.x]  += ssq[threadIdx.x + off];
        }
        __syncthreads();
    }
    if (threadIdx.x == 0) {
        float m = ssum[0] * (1.f / BATCH);
        float var = ssq[0] * (1.f / BATCH) - m * m;
        mu[f] = m;
        rstd[f] = rsqrtf(var + 1e-5f);
    }
}

// ---------------- K2: build normalized A matrix (2048 x 800) ----------------
__global__ void k_norm(const float* __restrict__ states,
                       const float* __restrict__ mu, const float* __restrict__ rstd,
                       float* __restrict__ An) {
    int idx = blockIdx.x * blockDim.x + threadIdx.x;   // b*800 + f
    if (idx >= BATCH * EHH) return;
    int b = idx / EHH, f = idx % EHH;
    int a = f >> 5, s = f & 31;
    float v = states[(size_t)a * BATCH * N_S + b * N_S + s];
    An[idx] = (v - mu[f]) * rstd[f];
}

// ---------------- K2b: LDS-tiled transpose of ehh_w (800x3200 -> 3200x800) ----------------
__global__ void __launch_bounds__(256) k_transpose(const float* __restrict__ src,
                                                   float* __restrict__ dst) {
    __shared__ float tile[32][33];
    int bx = blockIdx.x % (HID / 32);   // column tile: 0..99
    int by = blockIdx.x / (HID / 32);   // row tile:    0..24
    int tx = threadIdx.x & 31;
    int ty = threadIdx.x >> 5;          // 0..7
    for (int i = ty; i < 32; i += 8)
        tile[i][tx] = src[(size_t)(by * 32 + i) * HID + bx * 32 + tx];
    __syncthreads();
    for (int i = ty; i < 32; i += 8)
        dst[(size_t)(bx * 32 + i) * EHH + by * 32 + tx] = tile[tx][i];
}

// ---------------- K3: GEMM1  C(2048x3200) = leaky(A(2048x800) @ B(800x3200)) ----------------
// one wave computes a 32x64 tile; B consumed transposed (3200x800) so every
// fragment load is a single aligned b64. Double-buffered K loop: next k-step's
// fragments are in flight while the current 8 WMMAs issue.
__global__ void __launch_bounds__(256) k_gemm1(const float* __restrict__ A,
                                               const float* __restrict__ Bt,
                                               float* __restrict__ C) {
    int lane  = threadIdx.x & 31;
    int wave  = blockIdx.x * 8 + (threadIdx.x >> 5);
    int mTile = wave / (HID / 64);     // 0..63   (32-row tiles)
    int nSup  = wave % (HID / 64);     // 0..49
    int r16   = lane & 15;
    int khalf = lane >> 4;

    v8f acc[2][4];
#pragma unroll
    for (int mi = 0; mi < 2; mi++)
#pragma unroll
        for (int j = 0; j < 4; j++) acc[mi][j] = (v8f){};

    const float* arow0 = A + (size_t)(mTile * 32 + r16) * EHH;
    const float* arow1 = arow0 + (size_t)16 * EHH;
    int colBase = nSup * 64 + r16;
    const float* bcol[4];
#pragma unroll
    for (int j = 0; j < 4; j++) bcol[j] = Bt + (size_t)(colBase + j * 16) * EHH;

    int kk = 2 * khalf;
    v2f a0c = *(const v2f*)(arow0 + kk);
    v2f a1c = *(const v2f*)(arow1 + kk);
    v2f bc[4];
#pragma unroll
    for (int j = 0; j < 4; j++) bc[j] = *(const v2f*)(bcol[j] + kk);

    for (int k = 0; k < EHH; k += 4) {
        int kn = kk + 4;   // one-iteration lookahead (<=16B over-read lands in ws)
        __builtin_prefetch(&arow0[kn + 60], 0, 3);
        v2f a0n = *(const v2f*)(arow0 + kn);
        v2f a1n = *(const v2f*)(arow1 + kn);
        v2f bn[4];
#pragma unroll
        for (int j = 0; j < 4; j++) bn[j] = *(const v2f*)(bcol[j] + kn);
#pragma unroll
        for (int j = 0; j < 4; j++) {
            acc[0][j] = __builtin_amdgcn_wmma_f32_16x16x4_f32(
                false, a0c, false, bc[j], (short)0, acc[0][j], false, false);
            acc[1][j] = __builtin_amdgcn_wmma_f32_16x16x4_f32(
                false, a1c, false, bc[j], (short)0, acc[1][j], false, false);
        }
        a0c = a0n; a1c = a1n;
#pragma unroll
        for (int j = 0; j < 4; j++) bc[j] = bn[j];
        kk = kn;
    }

    int rbase = khalf * 8;
#pragma unroll
    for (int mi = 0; mi < 2; mi++)
#pragma unroll
        for (int j = 0; j < 4; j++) {
            int col = colBase + j * 16;
#pragma unroll
            for (int r = 0; r < 8; r++) {
                int row = mTile * 32 + mi * 16 + rbase + r;
                C[(size_t)row * HID + col] = leaky(acc[mi][j][r]);
            }
        }
}

// ---------------- K4: zero the neighbor-attention buffer ----------------
__global__ void k_zero(float* __restrict__ p, int n) {
    int i = blockIdx.x * blockDim.x + threadIdx.x;
    if (i < n) p[i] = 0.f;
}

// ---------------- K5: ordered edge scatter (single wave, serial over edges) ----------------
__global__ void k_scatter(const float* __restrict__ anova, const int* __restrict__ adj,
                          float* __restrict__ nb) {
    int a = threadIdx.x;
    if (a >= N_AGENT) return;
    const float* bi = anova + (size_t)HID * N_AGENT;   // rows [3200,3350) of anova
    for (int e = 0; e < N_EDGE; e++) {                 // phase 1: adj[:,1]
        float v = bi[adj[e * 4 + 0] * N_AGENT + a];
        nb[adj[e * 4 + 1] * N_AGENT + a] = v;
    }
    for (int e = 0; e < N_EDGE; e++) {                 // phase 2: adj[:,3] (overrides)
        float v = bi[adj[e * 4 + 0] * N_AGENT + a];
        nb[adj[e * 4 + 3] * N_AGENT + a] = v;
    }
}

// ---------------- K6: fold attention into w1 -> w1_eff, TRANSPOSED: (25 x 16 x 3200) ----------------
__global__ void k_w1eff(const float* __restrict__ anova, const float* __restrict__ nb,
                        const float* __restrict__ w1, float* __restrict__ W) {
    int idx = blockIdx.x * blockDim.x + threadIdx.x;   // a*3200 + h
    if (idx >= N_AGENT * HID) return;
    int a = idx / HID, h = idx % HID;
    float att = anova[h * N_AGENT + a] + nb[h * N_AGENT + a];
    const float* w1r = w1 + (size_t)(a * HID + h) * 12;
    float* wa = W + (size_t)a * 16 * HID;
#pragma unroll
    for (int k = 0; k < 12; k++) wa[(size_t)k * HID + h] = att * w1r[k];
#pragma unroll
    for (int k = 12; k < 16; k++) wa[(size_t)k * HID + h] = 0.f;
}

// ---------------- K7: GEMM2 per agent: h1 = leaky(E(2048x3200) @ w1_eff[a] + b1) ----------------
// one wave computes a 32x16 tile; w1_eff stored K-contiguous -> b64 loads.
// Double-buffered K loop like GEMM1.
__global__ void __launch_bounds__(256) k_gemm2(const float* __restrict__ E,
                                               const float* __restrict__ W,
                                               const float* __restrict__ b1,
                                               float* __restrict__ H1) {
    int lane  = threadIdx.x & 31;
    int wave  = blockIdx.x * 8 + (threadIdx.x >> 5);
    int a     = wave / (BATCH / 32);   // 0..24
    int mTile = wave % (BATCH / 32);   // 0..63
    int r16   = lane & 15;
    int khalf = lane >> 4;

    v8f acc0 = (v8f){};
    v8f acc1 = (v8f){};
    const float* arow0 = E + (size_t)(mTile * 32 + r16) * HID;
    const float* arow1 = arow0 + (size_t)16 * HID;
    const float* wrow  = W + (size_t)a * 16 * HID + (size_t)r16 * HID;

    int kk = 2 * khalf;
    v2f a0c = *(const v2f*)(arow0 + kk);
    v2f a1c = *(const v2f*)(arow1 + kk);
    v2f bfc = *(const v2f*)(wrow + kk);

    for (int k = 0; k < HID; k += 4) {
        int kn = kk + 4;   // lookahead (<=16B over-read lands in ws)
        __builtin_prefetch(&arow0[kn + 60], 0, 3);
        v2f a0n = *(const v2f*)(arow0 + kn);
        v2f a1n = *(const v2f*)(arow1 + kn);
        v2f bfn = *(const v2f*)(wrow + kn);
        acc0 = __builtin_amdgcn_wmma_f32_16x16x4_f32(
            false, a0c, false, bfc, (short)0, acc0, false, false);
        acc1 = __builtin_amdgcn_wmma_f32_16x16x4_f32(
            false, a1c, false, bfc, (short)0, acc1, false, false);
        a0c = a0n; a1c = a1n; bfc = bfn;
        kk = kn;
    }

    float bias = (r16 < 12) ? b1[a * 12 + r16] : 0.f;
    int rbase = khalf * 8;
#pragma unroll
    for (int r = 0; r < 8; r++) {
        int row0 = mTile * 32 + rbase + r;
        int row1 = row0 + 16;
        H1[(size_t)a * BATCH * 16 + (size_t)row0 * 16 + r16] = leaky(acc0[r] + bias);
        H1[(size_t)a * BATCH * 16 + (size_t)row1 * 16 + r16] = leaky(acc1[r] + bias);
    }
}

// ---------------- K8: tiny MLP head + action gather -> q (25 x 2048) ----------------
__global__ void k_head(const float* __restrict__ H1,
                       const float* __restrict__ w2, const float* __restrict__ b2,
                       const float* __restrict__ w3, const float* __restrict__ b3,
                       const int* __restrict__ actions, float* __restrict__ out) {
    int idx = blockIdx.x * blockDim.x + threadIdx.x;   // a*2048 + b
    if (idx >= N_AGENT * BATCH) return;
    int a = idx / BATCH, b = idx % BATCH;
    const float* h1 = H1 + (size_t)(a * BATCH + b) * 16;
    float h2[12];
#pragma unroll
    for (int j = 0; j < 12; j++) {
        float s = b2[a * 12 + j];
#pragma unroll
        for (int k = 0; k < 12; k++) s += h1[k] * w2[(size_t)(a * 12 + k) * 12 + j];
        h2[j] = leaky(s);
    }
    int act = actions[a * BATCH + b];
    float s = b3[a * N_A + act];
#pragma unroll
    for (int j = 0; j < 12; j++) s += h2[j] * w3[(size_t)(a * 12 + j) * N_A + act];
    out[(size_t)a * BATCH + b] = s;
}

extern "C" void kernel_launch(void* const* d_in, const int* in_sizes, int n_in,
                              void* d_out, int out_size, void* d_ws, size_t ws_size,
                              hipStream_t stream) {
    const float* states  = (const float*)d_in[0];
    const float* ehh_w   = (const float*)d_in[1];
    const float* anova   = (const float*)d_in[2];
    const float* w1      = (const float*)d_in[3];
    const float* b1      = (const float*)d_in[4];
    const float* w2      = (const float*)d_in[5];
    const float* b2      = (const float*)d_in[6];
    const float* w3      = (const float*)d_in[7];
    const float* b3      = (const float*)d_in[8];
    const int*   actions = (const int*)d_in[9];
    const int*   adj     = (const int*)d_in[10];
    float* out = (float*)d_out;

    float* ws      = (float*)d_ws;
    float* mu      = ws;                                   // 800
    float* rstd    = mu + 800;                             // 800
    float* ehh_n   = rstd + 800;                           // 2048*800
    float* ehh_wt  = ehh_n + (size_t)BATCH * EHH;          // 3200*800 (transposed weights)
    float* ehh_emb = ehh_wt + (size_t)HID * EHH;           // 2048*3200
    float* nb      = ehh_emb + (size_t)BATCH * HID;        // 3200*25
    float* w1eff   = nb + (size_t)HID * N_AGENT;           // 25*16*3200
    float* h1      = w1eff + (size_t)N_AGENT * 16 * HID;   // 25*2048*16

    k_meanvar<<<EHH, 256, 0, stream>>>(states, mu, rstd);
    k_norm<<<(BATCH * EHH + 255) / 256, 256, 0, stream>>>(states, mu, rstd, ehh_n);
    k_transpose<<<(HID / 32) * (EHH / 32), 256, 0, stream>>>(ehh_w, ehh_wt);
    k_gemm1<<<(64 * (HID / 64)) / 8, 256, 0, stream>>>(ehh_n, ehh_wt, ehh_emb);
    k_zero<<<(HID * N_AGENT + 255) / 256, 256, 0, stream>>>(nb, HID * N_AGENT);
    k_scatter<<<1, 32, 0, stream>>>(anova, adj, nb);
    k_w1eff<<<(N_AGENT * HID + 255) / 256, 256, 0, stream>>>(anova, nb, w1, w1eff);
    k_gemm2<<<(N_AGENT * (BATCH / 32)) / 8, 256, 0, stream>>>(ehh_emb, w1eff, b1, h1);
    k_head<<<(N_AGENT * BATCH + 255) / 256, 256, 0, stream>>>(h1, w2, b2, w3, b3, actions, out);
}